// H2RDetector_8512625180955
// MI455X (gfx1250) — compile-verified
//
#include <hip/hip_runtime.h>
#include <math.h>
#include <stdint.h>

#define B_   32
#define H_   512
#define W_   512
#define HW_  (H_ * W_)
#define K_   1000
#define CAP_ 32768          // max candidates kept per image (expected ~29k peaks)

#define TW 64               // tile width  (interior)
#define TH 16               // tile height (interior)
#define HALO_W (TW + 2)     // 66
#define HALO_H (TH + 2)     // 18
#define PITCH  HALO_W       // contiguous rows: TDM fill == manual fill layout

typedef uint32_t v4u __attribute__((ext_vector_type(4)));
typedef int      v4i __attribute__((ext_vector_type(4)));
typedef int      v8i __attribute__((ext_vector_type(8)));

#if __has_builtin(__builtin_amdgcn_tensor_load_to_lds)
#define HAVE_TDM 1
#else
#define HAVE_TDM 0
#endif

__device__ __forceinline__ float sigmoidf_(float x) {
  return 1.0f / (1.0f + __expf(-x));
}

// ---------------------------------------------------------------------------
// Pass 0: zero the per-batch candidate counters (graph-safe, deterministic)
// ---------------------------------------------------------------------------
__global__ void zero_counters_kernel(int* __restrict__ counters) {
  if (threadIdx.x < B_) counters[threadIdx.x] = 0;
}

#if HAVE_TDM
// Issue one TDM 2D tile load: (HALO_W x HALO_H) f32 tile at (gx0,gy0) of a
// 512x512 tensor -> contiguous LDS at lds_addr. Wave-level op (EXEC ignored),
// tracked on TENSORcnt.
__device__ __forceinline__ void tdm_load_tile(uint64_t gbase, int gx0, int gy0,
                                              uint32_t lds_addr) {
  const uint64_t ga = gbase + (uint64_t)((gy0 * W_ + gx0) * 4);
  v4u g0;
  g0[0] = 1u;                                   // count=1, user descriptor
  g0[1] = lds_addr;                             // LDS byte address
  g0[2] = (uint32_t)ga;                         // global_addr[31:0]
  g0[3] = (uint32_t)(ga >> 32) | 0x80000000u;   // global_addr[56:32] | type=2
  v8i g1;
  g1[0] = 0x20000;                              // data_size=2 (4B), mask=0
  g1[1] = (int)(512u << 16);                    // tensor_dim0 = 512 (lo16)
  g1[2] = (int)(512u << 16);                    // tensor_dim1 = 512 (lo16)
  g1[3] = (int)((uint32_t)HALO_W << 16);        // tile_dim0 = 66
  g1[4] = HALO_H;                               // tile_dim1 = 18, tile_dim2 = 0
  g1[5] = 512;                                  // tensor_dim0_stride[31:0]
  g1[6] = 0;                                    // stride hi / dim1_stride lo
  g1[7] = 0;
  v4i gz4 = {0, 0, 0, 0};
#if __clang_major__ >= 23
  v8i gz8 = {0, 0, 0, 0, 0, 0, 0, 0};
  __builtin_amdgcn_tensor_load_to_lds(g0, g1, gz4, gz4, gz8, 0);
#else
  __builtin_amdgcn_tensor_load_to_lds(g0, g1, gz4, gz4, 0);
#endif
}
#endif

// ---------------------------------------------------------------------------
// Pass A: fused score + 3x3 NMS + candidate compaction.
// Interior tiles: halo staged by the Tensor Data Mover (TENSORcnt).
// Border tiles:   per-lane async global->LDS loads with clamping (ASYNCcnt).
// ---------------------------------------------------------------------------
__global__ __launch_bounds__(256) void score_nms_kernel(
    const float* __restrict__ route,
    const float* __restrict__ unc,
    int*   __restrict__ counters,
    float* __restrict__ candS,
    int*   __restrict__ candI)
{
  __shared__ float shR[HALO_H * PITCH];
  __shared__ float shU[HALO_H * PITCH];
  __shared__ float shS[HALO_H * PITCH];

  const int tile = blockIdx.x;          // B_ * 256 tiles
  const int b    = tile >> 8;           // 256 tiles per image (8 x 32)
  const int t    = tile & 255;
  const int ty   = t >> 3;              // tiles_x = 512/64 = 8
  const int tx   = t & 7;
  const int x0   = tx * TW;
  const int y0   = ty * TH;
  const int tid  = threadIdx.x;

  const float* rbase = route + (size_t)b * HW_;
  const float* ubase = unc   + (size_t)b * HW_;
  const uint64_t rb64 = (uint64_t)(uintptr_t)rbase;
  const uint64_t ub64 = (uint64_t)(uintptr_t)ubase;

#if HAVE_TDM
  const bool interior = (tx > 0) && (tx < 7) && (ty > 0) && (ty < 31);
#else
  const bool interior = false;
#endif

  if (interior) {
#if HAVE_TDM
    // one DMA per tensor per workgroup: only wave 0 issues & drains
    if (tid < 32) {
      tdm_load_tile(rb64, x0 - 1, y0 - 1, (uint32_t)(uintptr_t)&shR[0]);
      tdm_load_tile(ub64, x0 - 1, y0 - 1, (uint32_t)(uintptr_t)&shU[0]);
      __builtin_amdgcn_s_wait_tensorcnt(0);
    }
    __syncthreads();
#endif
  } else {
    // ---- stage the 66x18 halo via gfx1250 async global->LDS loads ----
    for (int i = tid; i < HALO_H * HALO_W; i += 256) {
      const int hy  = i / HALO_W;
      const int hx  = i - hy * HALO_W;
      const int gy  = y0 - 1 + hy;
      const int gx  = x0 - 1 + hx;
      const int cgy = min(max(gy, 0), H_ - 1);   // clamp address; OOB slots are
      const int cgx = min(max(gx, 0), W_ - 1);   // overwritten with -inf below
      const uint32_t goff = (uint32_t)((cgy * W_ + cgx) * 4);
      const uint32_t ldsR = (uint32_t)(uintptr_t)&shR[hy * PITCH + hx];
      const uint32_t ldsU = (uint32_t)(uintptr_t)&shU[hy * PITCH + hx];
      asm volatile("global_load_async_to_lds_b32 %0, %1, %2"
                   :: "v"(ldsR), "v"(goff), "s"(rb64) : "memory");
      asm volatile("global_load_async_to_lds_b32 %0, %1, %2"
                   :: "v"(ldsU), "v"(goff), "s"(ub64) : "memory");
    }
    asm volatile("s_wait_asynccnt 0x0" ::: "memory");
    __syncthreads();
  }

  // ---- build the score map in LDS (OOB halo = -inf, mirrors SAME/-inf pad) ----
  for (int i = tid; i < HALO_H * HALO_W; i += 256) {
    const int hy = i / HALO_W;
    const int hx = i - hy * HALO_W;
    const int gy = y0 - 1 + hy;
    const int gx = x0 - 1 + hx;
    float s;
    if (gy < 0 || gy >= H_ || gx < 0 || gx >= W_) {
      s = -3.0e38f;
    } else {
      const float sr = sigmoidf_(shR[hy * PITCH + hx]);
      const float su = sigmoidf_(shU[hy * PITCH + hx]);
      s = sr * sr * (1.0f - 0.35f * su);       // sigmoid^2 * (1 - w*sig(unc))
    }
    shS[hy * PITCH + hx] = s;
  }
  __syncthreads();

  // ---- 3x3 peak test on interior pixels; compact peaks with atomics ----
  for (int i = tid; i < TW * TH; i += 256) {
    const int py = i >> 6;          // /64
    const int px = i & 63;
    const int ly = py + 1;
    const int lx = px + 1;
    const float s = shS[ly * PITCH + lx];
    float m;
    m =            shS[(ly - 1) * PITCH + (lx - 1)];
    m = fmaxf(m,   shS[(ly - 1) * PITCH + (lx    )]);
    m = fmaxf(m,   shS[(ly - 1) * PITCH + (lx + 1)]);
    m = fmaxf(m,   shS[(ly    ) * PITCH + (lx - 1)]);
    m = fmaxf(m,   shS[(ly    ) * PITCH + (lx + 1)]);
    m = fmaxf(m,   shS[(ly + 1) * PITCH + (lx - 1)]);
    m = fmaxf(m,   shS[(ly + 1) * PITCH + (lx    )]);
    m = fmaxf(m,   shS[(ly + 1) * PITCH + (lx + 1)]);
    if (s >= m) {                   // score == 3x3 pooled max
      const int pos = atomicAdd(&counters[b], 1);
      if (pos < CAP_) {
        candS[(size_t)b * CAP_ + pos] = s;
        candI[(size_t)b * CAP_ + pos] = (y0 + py) * W_ + (x0 + px);
      }
    }
  }
}

// ---------------------------------------------------------------------------
// Pass B: per-image exact top-K (MSB radix select on positive-float bits),
// bitonic sort of the K winners, ROI construction + padded outputs.
// ---------------------------------------------------------------------------
__global__ __launch_bounds__(1024) void topk_rois_kernel(
    const float* __restrict__ scale,
    const float* __restrict__ unc,
    const int*   __restrict__ counters,
    const float* __restrict__ candS,
    const int*   __restrict__ candI,
    const int*   __restrict__ p_ih,
    const int*   __restrict__ p_iw,
    float* __restrict__ out)
{
  __shared__ unsigned hist[256];
  __shared__ float    topS[1024];
  __shared__ int      topI[1024];
  __shared__ unsigned sh_prefix, sh_remK, sh_cntA, sh_cntEq;

  const int b   = blockIdx.x;
  const int tid = threadIdx.x;

  int n = counters[b];
  if (n > CAP_) n = CAP_;
  const float* cs = candS + (size_t)b * CAP_;
  const int*   ci = candI + (size_t)b * CAP_;

  topS[tid] = -3.0e38f;
  topI[tid] = -1;
  if (tid == 0) { sh_cntA = 0u; sh_cntEq = 0u; }
  __syncthreads();

  if (n <= K_) {
    // fewer candidates than K: take everything
    for (int i = tid; i < n; i += 1024) {
      const unsigned p = atomicAdd(&sh_cntA, 1u);
      topS[p] = cs[i];
      topI[p] = ci[i];
    }
    __syncthreads();
  } else {
    // ---- 4-digit MSB radix select (scores > 0 => bits are order-preserving) ----
    unsigned prefix = 0u, highMask = 0u, remK = K_;
    for (int shift = 24; shift >= 0; shift -= 8) {
      if (tid < 256) hist[tid] = 0u;
      __syncthreads();
      for (int i = tid; i < n; i += 1024) {
        const unsigned u = __float_as_uint(cs[i]);
        if ((u & highMask) == prefix)
          atomicAdd(&hist[(u >> shift) & 255u], 1u);
      }
      __syncthreads();
      if (tid == 0) {
        unsigned cum = 0u, g = 0u, rk = remK;
        for (int d = 255; d >= 0; --d) {
          const unsigned c = hist[d];
          if (cum + c >= rk) { g = (unsigned)d; rk -= cum; break; }
          cum += c;
        }
        sh_prefix = prefix | (g << shift);
        sh_remK   = rk;
      }
      __syncthreads();
      prefix    = sh_prefix;
      remK      = sh_remK;
      highMask |= (0xFFu << shift);
      __syncthreads();
    }
    const unsigned T      = prefix;        // bits of the K-th largest score
    const unsigned needEq = remK;          // how many ties at T to keep
    const unsigned gtBase = (unsigned)K_ - needEq;

    for (int i = tid; i < n; i += 1024) {
      const unsigned u = __float_as_uint(cs[i]);
      if (u > T) {
        const unsigned p = atomicAdd(&sh_cntA, 1u);   // exactly gtBase of these
        topS[p] = cs[i];
        topI[p] = ci[i];
      } else if (u == T) {
        const unsigned e = atomicAdd(&sh_cntEq, 1u);
        if (e < needEq) {
          topS[gtBase + e] = cs[i];
          topI[gtBase + e] = ci[i];
        }
      }
    }
    __syncthreads();
  }

  // ---- bitonic sort 1024 entries, descending by score (pads sink to end) ----
  for (unsigned k = 2; k <= 1024; k <<= 1) {
    for (unsigned j = k >> 1; j > 0; j >>= 1) {
      const unsigned i   = (unsigned)tid;
      const unsigned ixj = i ^ j;
      if (ixj > i) {
        const float si = topS[i], sj = topS[ixj];
        const bool  up = ((i & k) == 0);
        const bool  sw = up ? (si < sj) : (si > sj);
        if (sw) {
          const int ii = topI[i], ij = topI[ixj];
          topS[i] = sj; topS[ixj] = si;
          topI[i] = ij; topI[ixj] = ii;
        }
      }
      __syncthreads();
    }
  }

  // ---- ROI construction + padded outputs ----
  const float ihf = (float)p_ih[0];
  const float iwf = (float)p_iw[0];
  float* rois = out;
  float* scso = out + (size_t)B_ * K_ * 5;
  float* vald = out + (size_t)B_ * K_ * 5 + (size_t)B_ * K_;

  if (tid < K_) {
    const float s   = topS[tid];
    const int   idx = topI[tid];
    float* r = rois + ((size_t)b * K_ + tid) * 5;
    if (idx >= 0) {
      const int y = idx / W_;
      const int x = idx - y * W_;
      const float cx = ((float)x + 0.5f) * 4.0f;
      const float cy = ((float)y + 0.5f) * 4.0f;
      const float sc = scale[(size_t)b * HW_ + idx];
      const float ul = unc  [(size_t)b * HW_ + idx];
      const float su = sigmoidf_(sc);
      const float uu = sigmoidf_(ul);
      const float side = (32.0f + su * (512.0f - 32.0f)) * (1.0f + 0.25f * uu);
      const float half = 0.5f * side;
      const float x1 = fminf(fmaxf(cx - half, 0.0f), iwf - 1.0f);
      const float y1 = fminf(fmaxf(cy - half, 0.0f), ihf - 1.0f);
      const float x2 = fminf(fmaxf(cx + half, 1.0f), iwf);
      const float y2 = fminf(fmaxf(cy + half, 1.0f), ihf);
      r[0] = (float)b; r[1] = x1; r[2] = y1; r[3] = x2; r[4] = y2;
      scso[(size_t)b * K_ + tid] = s;
      vald[(size_t)b * K_ + tid] = 1.0f;
    } else {
      r[0] = 0.0f; r[1] = 0.0f; r[2] = 0.0f; r[3] = 0.0f; r[4] = 0.0f;
      scso[(size_t)b * K_ + tid] = 0.0f;
      vald[(size_t)b * K_ + tid] = 0.0f;
    }
  }
}

// ---------------------------------------------------------------------------
extern "C" void kernel_launch(void* const* d_in, const int* in_sizes, int n_in,
                              void* d_out, int out_size, void* d_ws, size_t ws_size,
                              hipStream_t stream) {
  const float* route = (const float*)d_in[0];
  const float* scale = (const float*)d_in[1];
  const float* unc   = (const float*)d_in[2];
  const int*   ih    = (const int*)d_in[3];
  const int*   iw    = (const int*)d_in[4];
  float* out = (float*)d_out;

  // workspace layout: [32 counters][B*CAP scores][B*CAP indices]  (~8.4 MB)
  int*   counters = (int*)d_ws;
  float* candS    = (float*)((char*)d_ws + 256);
  int*   candI    = (int*)  ((char*)d_ws + 256 + (size_t)B_ * CAP_ * sizeof(float));

  zero_counters_kernel<<<1, 32, 0, stream>>>(counters);
  score_nms_kernel<<<B_ * 256, 256, 0, stream>>>(route, unc, counters, candS, candI);
  topk_rois_kernel<<<B_, 1024, 0, stream>>>(scale, unc, counters, candS, candI, ih, iw, out);
}